// BlockSparse_68994354643218
// MI455X (gfx1250) — compile-verified
//
#include <hip/hip_runtime.h>
#include <stdint.h>

typedef __attribute__((ext_vector_type(2))) float v2f;
typedef __attribute__((ext_vector_type(8))) float v8f;

#define MATSIZE 4096
#define BATCH   8192
#define BLK     256     // mask block size
#define NBLK    16      // blocks per row
#define BM      128
#define BN      128
#define KC      16      // K-chunk staged per buffer
#define LDW     20      // LDS row stride (floats): 80B rows -> 16B aligned, conflict-free
#define TILE_F  (BM * LDW)

__global__ __launch_bounds__(256)
void bsmm_f32_wmma(const float* __restrict__ x,
                   const float* __restrict__ w,
                   const float* __restrict__ bias,
                   const uint8_t* __restrict__ mask,
                   float* __restrict__ out) {
  __shared__ float As[2][TILE_F];   // x tiles:  2 x (128 x 16)
  __shared__ float Bs[2][TILE_F];   // w tiles:  2 x (128 x 16)

  const int tid  = threadIdx.x;
  const int lane = tid & 31;
  const int wave = tid >> 5;
  const int lo   = lane & 15;
  const int hi   = lane >> 4;

  const int n_base = blockIdx.x * BN;
  const int m_base = blockIdx.y * BM;

  const int wm = wave & 1;    // 2 wave-rows  -> 64 output rows each
  const int wn = wave >> 1;   // 4 wave-cols  -> 32 output cols each

  v8f zero = {};
  v8f acc[4][2];
#pragma unroll
  for (int i = 0; i < 4; ++i)
#pragma unroll
    for (int j = 0; j < 2; ++j) acc[i][j] = zero;

  // One mask byte per 256x256 block (uniform across workgroup).
  unsigned active = 0;
#pragma unroll
  for (int kb = 0; kb < NBLK; ++kb) {
    if (mask[(size_t)n_base * MATSIZE + (size_t)kb * BLK]) active |= (1u << kb);
  }
  int kbs[NBLK];
  int nact = 0;
#pragma unroll
  for (int kb = 0; kb < NBLK; ++kb) {
    if (active & (1u << kb)) kbs[nact++] = kb;
  }
  const int nch = nact * (BLK / KC);   // 16 chunks per active block

  // Per-thread staging geometry: 128x16 floats = 512 b128 per matrix,
  // 256 threads -> 2 async b128 per thread per matrix per chunk.
  const int row0 = tid >> 2;           // i=0 row
  const int seg0 = tid & 3;
  const int row1 = (tid + 256) >> 2;   // i=1 row
  const int seg1 = (tid + 256) & 3;

  // Async global->LDS prefetch of one K-chunk into buffer `buf`.
  auto prefetch = [&](int buf, int k0) {
    {
      const uint32_t lA = (uint32_t)(uintptr_t)&As[buf][row0 * LDW + seg0 * 4];
      const uint64_t gA = (uint64_t)(uintptr_t)(x + (size_t)(m_base + row0) * MATSIZE + k0 + seg0 * 4);
      asm volatile("global_load_async_to_lds_b128 %0, %1, off" :: "v"(lA), "v"(gA) : "memory");
      const uint32_t lB = (uint32_t)(uintptr_t)&Bs[buf][row0 * LDW + seg0 * 4];
      const uint64_t gB = (uint64_t)(uintptr_t)(w + (size_t)(n_base + row0) * MATSIZE + k0 + seg0 * 4);
      asm volatile("global_load_async_to_lds_b128 %0, %1, off" :: "v"(lB), "v"(gB) : "memory");
    }
    {
      const uint32_t lA = (uint32_t)(uintptr_t)&As[buf][row1 * LDW + seg1 * 4];
      const uint64_t gA = (uint64_t)(uintptr_t)(x + (size_t)(m_base + row1) * MATSIZE + k0 + seg1 * 4);
      asm volatile("global_load_async_to_lds_b128 %0, %1, off" :: "v"(lA), "v"(gA) : "memory");
      const uint32_t lB = (uint32_t)(uintptr_t)&Bs[buf][row1 * LDW + seg1 * 4];
      const uint64_t gB = (uint64_t)(uintptr_t)(w + (size_t)(n_base + row1) * MATSIZE + k0 + seg1 * 4);
      asm volatile("global_load_async_to_lds_b128 %0, %1, off" :: "v"(lB), "v"(gB) : "memory");
    }
  };

  auto k_of = [&](int t) { return kbs[t >> 4] * BLK + (t & 15) * KC; };

  if (nch > 0) prefetch(0, k_of(0));

  for (int t = 0; t < nch; ++t) {
    const int cur = t & 1;

    // Issue next chunk's DMA before computing the current one.
    if (t + 1 < nch) {
      prefetch(cur ^ 1, k_of(t + 1));
      // 4 of my async ops belong to the next chunk; current chunk complete
      // once ASYNCcnt drops to 4 (in-order completion per wave).
      asm volatile("s_wait_asynccnt 0x4" ::: "memory");
    } else {
      asm volatile("s_wait_asynccnt 0x0" ::: "memory");
    }
    __syncthreads();   // current buffer fully resident for all waves

    // 4 k-steps of K=4; 4x2 WMMA tiles per step -> 32 WMMA per chunk.
#pragma unroll
    for (int kk = 0; kk < KC / 4; ++kk) {
      const int kcol = kk * 4 + 2 * hi;   // even -> 8B-aligned ds_load_b64
      v2f a[4], b[2];
#pragma unroll
      for (int mt = 0; mt < 4; ++mt) {
        const int r = wm * 64 + mt * 16 + lo;
        a[mt] = *reinterpret_cast<const v2f*>(&As[cur][r * LDW + kcol]);
      }
#pragma unroll
      for (int nt = 0; nt < 2; ++nt) {
        const int r = wn * 32 + nt * 16 + lo;
        b[nt] = *reinterpret_cast<const v2f*>(&Bs[cur][r * LDW + kcol]);
      }
#pragma unroll
      for (int mt = 0; mt < 4; ++mt)
#pragma unroll
        for (int nt = 0; nt < 2; ++nt)
          acc[mt][nt] = __builtin_amdgcn_wmma_f32_16x16x4_f32(
              false, a[mt], false, b[nt], (short)0, acc[mt][nt], false, false);
    }

    __syncthreads();   // all reads of buf[cur] done before t+2's prefetch overwrites it
  }

  // Epilogue: C/D layout -> lane covers N = n_tile+lo, rows M = base + 8*hi + r.
#pragma unroll
  for (int nt = 0; nt < 2; ++nt) {
    const int n  = n_base + wn * 32 + nt * 16 + lo;
    const float bv = bias[n];
#pragma unroll
    for (int mt = 0; mt < 4; ++mt) {
      const int m0 = m_base + wm * 64 + mt * 16 + hi * 8;
#pragma unroll
      for (int r = 0; r < 8; ++r) {
        out[(size_t)(m0 + r) * MATSIZE + n] = acc[mt][nt][r] + bv;
      }
    }
  }
}

extern "C" void kernel_launch(void* const* d_in, const int* in_sizes, int n_in,
                              void* d_out, int out_size, void* d_ws, size_t ws_size,
                              hipStream_t stream) {
  const float*   x    = (const float*)d_in[0];
  const float*   w    = (const float*)d_in[1];
  const float*   bias = (const float*)d_in[2];
  const uint8_t* mask = (const uint8_t*)d_in[3];
  float* out = (float*)d_out;

  dim3 grid(MATSIZE / BN, BATCH / BM);   // 32 x 64 workgroups
  dim3 block(256);                        // 8 waves (wave32)
  bsmm_f32_wmma<<<grid, block, 0, stream>>>(x, w, bias, mask, out);
}